// TalkerAttentionV2_90864328114926
// MI455X (gfx1250) — compile-verified
//
#include <hip/hip_runtime.h>
#include <hip/hip_bf16.h>
#include <math.h>

// Problem constants (match reference)
#define B_    16
#define HID_  1024
#define NH_   16
#define NKV_  8
#define HD_   128
#define S_    4096
#define DKV_  1024            // NKV*HD
#define SCALE_ 0.08838834764831845f   // 128^-0.5
#define EPS_  1e-6f

typedef float v2f __attribute__((ext_vector_type(2)));
typedef float v4f __attribute__((ext_vector_type(4)));
typedef float v8f __attribute__((ext_vector_type(8)));

// ---------------------------------------------------------------------------
// WMMA f32 16x16x4 tile GEMM helper.
// A (16xK): rows = 16 batches. Per ISA layout: lanes 0-15 hold M=lane with
// K = kb+0 / kb+1 in the two A VGPRs, lanes 16-31 hold the same M with
// K = kb+2 / kb+3  -> per-lane v2f = { xrow[kb+2*hi], xrow[kb+2*hi+1] }.
// B (Kx16): B[k][n] = W[n][k]; mirrored layout -> per-lane v2f from W row n.
// ---------------------------------------------------------------------------
__device__ inline v8f gemm_tile_f32(const float* __restrict__ xrow,
                                    const float* __restrict__ wrow, int K) {
  const int lane = threadIdx.x & 31;
  const int hi   = lane >> 4;
  v8f c = {0.f, 0.f, 0.f, 0.f, 0.f, 0.f, 0.f, 0.f};
  for (int kb = 0; kb < K; kb += 4) {
    const int kk = kb + 2 * hi;
    v2f a;  a.x  = xrow[kk]; a.y  = xrow[kk + 1];
    v2f bm; bm.x = wrow[kk]; bm.y = wrow[kk + 1];
    // 8 args: (neg_a, A, neg_b, B, c_mod, C, reuse_a, reuse_b)
    c = __builtin_amdgcn_wmma_f32_16x16x4_f32(false, a, false, bm,
                                              (short)0, c, false, false);
  }
  return c;
}

// ---------------------------------------------------------------------------
// Kernel 1: fused QKV projection.  qkv[b][n] = sum_k x[b][k] * W[n][k]
// n in [0,2048) -> Wq ; [2048,3072) -> Wk ; [3072,4096) -> Wv
// One wave (32 threads) per 16x16 output tile; 256 tiles.
// ---------------------------------------------------------------------------
__global__ __launch_bounds__(32) void qkv_proj_kernel(
    const float* __restrict__ x,  const float* __restrict__ Wq,
    const float* __restrict__ Wk, const float* __restrict__ Wv,
    float* __restrict__ qkv) {
  const int lane = threadIdx.x & 31;
  const int row  = lane & 15;          // batch row / B-matrix column
  const int hi   = lane >> 4;
  const int n0   = blockIdx.x * 16;
  const int n    = n0 + row;

  const float* W; int nr;
  if (n < 2048)      { W = Wq; nr = n; }
  else if (n < 3072) { W = Wk; nr = n - 2048; }
  else               { W = Wv; nr = n - 3072; }

  const float* wrow = W + (size_t)nr * HID_;
  const float* xrow = x + (size_t)row * HID_;

  v8f c = gemm_tile_f32(xrow, wrow, HID_);

  // D layout: VGPR r -> M = r + 8*hi, N = n0 + (lane&15)
  for (int r = 0; r < 8; ++r) {
    const int m = r + 8 * hi;
    qkv[(size_t)m * 4096 + n0 + row] = c[r];
  }
}

// ---------------------------------------------------------------------------
// Kernel 2: per-head RMSNorm + RoPE for q (16 heads) and k (8 heads).
// grid = B * 24 blocks, 128 threads each.
// ---------------------------------------------------------------------------
__global__ __launch_bounds__(128) void norm_rope_kernel(
    const float* __restrict__ qkv, const float* __restrict__ qnw,
    const float* __restrict__ knw, const float* __restrict__ rc,
    const float* __restrict__ rs,  float* __restrict__ qf,
    float* __restrict__ kf) {
  const int b  = blockIdx.x / 24;
  const int hh = blockIdx.x % 24;
  const int t  = threadIdx.x;

  __shared__ float red[128];
  __shared__ float nrm[128];

  const float* src; const float* w; float* dst;
  if (hh < 16) {
    src = qkv + (size_t)b * 4096 + hh * 128;
    w = qnw;  dst = qf + (size_t)b * 2048 + hh * 128;
  } else {
    src = qkv + (size_t)b * 4096 + 2048 + (hh - 16) * 128;
    w = knw;  dst = kf + (size_t)b * 1024 + (hh - 16) * 128;
  }

  const float v = src[t];
  red[t] = v * v;
  __syncthreads();
  for (int off = 64; off >= 1; off >>= 1) {
    if (t < off) red[t] += red[t + off];
    __syncthreads();
  }
  const float norm = rsqrtf(red[0] * (1.0f / 128.0f) + EPS_);
  const float tn   = v * norm * w[t];
  nrm[t] = tn;
  __syncthreads();
  const float rot = (t < 64) ? -nrm[t + 64] : nrm[t - 64];
  dst[t] = tn * rc[(size_t)b * 128 + t] + rot * rs[(size_t)b * 128 + t];
}

// wave32 reductions
__device__ inline float wave_max(float v) {
  for (int off = 16; off >= 1; off >>= 1)
    v = fmaxf(v, __shfl_xor(v, off, 32));
  return v;
}
__device__ inline float wave_sum(float v) {
  for (int off = 16; off >= 1; off >>= 1)
    v += __shfl_xor(v, off, 32);
  return v;
}

// ---------------------------------------------------------------------------
// Kernel 3: fused KV-cache blend/store + GQA attention.
// One workgroup per (b, kv-head); handles both query heads of the group.
// 256 threads; thread t owns s = j*1024 + 4*t + c (j=0..3, c=0..3):
// fully-coalesced b128 loads/stores, non-temporal (streamed once).
// K-phase: blend old K cache with new k, store new cache, accumulate scores.
// Softmax across workgroup. V-phase: blend/store V cache, accumulate attn*V.
// ---------------------------------------------------------------------------
__global__ __launch_bounds__(256) void attn_fused_kernel(
    const float* __restrict__ qf,    const float* __restrict__ kf,
    const float* __restrict__ qkv,   const float* __restrict__ um,
    const float* __restrict__ kpm,
    const float* __restrict__ kc_in, const float* __restrict__ vc_in,
    float* __restrict__ kc_out,      float* __restrict__ vc_out,
    float* __restrict__ oh) {
  const int b    = blockIdx.x / NKV_;
  const int kvh  = blockIdx.x % NKV_;
  const int t    = threadIdx.x;
  const int lane = t & 31;
  const int wid  = t >> 5;
  const int h0   = 2 * kvh;

  __shared__ float q0s[HD_], q1s[HD_], kvs[HD_], vvs[HD_];
  __shared__ float o0s[HD_], o1s[HD_];
  __shared__ float red0[8], red1[8];

  if (t < HD_) {
    q0s[t] = qf[(size_t)b * 2048 + h0 * HD_ + t];
    q1s[t] = qf[(size_t)b * 2048 + (h0 + 1) * HD_ + t];
    kvs[t] = kf[(size_t)b * 1024 + kvh * HD_ + t];
    vvs[t] = qkv[(size_t)b * 4096 + 3072 + kvh * HD_ + t];
    o0s[t] = 0.f; o1s[t] = 0.f;
  }

  // update mask for this thread's 16 s-positions, as 4 float4s
  const v4f* um4 = (const v4f*)(um + (size_t)b * S_);
  v4f umr[4];
#pragma unroll
  for (int j = 0; j < 4; ++j) umr[j] = um4[j * 256 + t];
  __syncthreads();

  float sc0[16], sc1[16];
#pragma unroll
  for (int j = 0; j < 16; ++j) { sc0[j] = 0.f; sc1[j] = 0.f; }

  // ---- Phase 1: K-cache blend + store (NT, b128) + scores ---------------
  for (int d = 0; d < HD_; ++d) {
    const size_t rowoff = ((size_t)b * DKV_ + kvh * HD_ + d) * S_;
    const v4f* src4 = (const v4f*)(kc_in + rowoff);
    v4f*       dst4 = (v4f*)(kc_out + rowoff);
    if (d + 1 < HD_) __builtin_prefetch(src4 + 1024 + t, 0, 1);
    const float kd = kvs[d], qc0 = q0s[d], qc1 = q1s[d];
#pragma unroll
    for (int j = 0; j < 4; ++j) {
      const v4f o = __builtin_nontemporal_load(src4 + j * 256 + t);
      const v4f m = umr[j];
      v4f nv;
      nv.x = fmaf(kd - o.x, m.x, o.x);
      nv.y = fmaf(kd - o.y, m.y, o.y);
      nv.z = fmaf(kd - o.z, m.z, o.z);
      nv.w = fmaf(kd - o.w, m.w, o.w);
      __builtin_nontemporal_store(nv, dst4 + j * 256 + t);
      sc0[j * 4 + 0] = fmaf(qc0, nv.x, sc0[j * 4 + 0]);
      sc0[j * 4 + 1] = fmaf(qc0, nv.y, sc0[j * 4 + 1]);
      sc0[j * 4 + 2] = fmaf(qc0, nv.z, sc0[j * 4 + 2]);
      sc0[j * 4 + 3] = fmaf(qc0, nv.w, sc0[j * 4 + 3]);
      sc1[j * 4 + 0] = fmaf(qc1, nv.x, sc1[j * 4 + 0]);
      sc1[j * 4 + 1] = fmaf(qc1, nv.y, sc1[j * 4 + 1]);
      sc1[j * 4 + 2] = fmaf(qc1, nv.z, sc1[j * 4 + 2]);
      sc1[j * 4 + 3] = fmaf(qc1, nv.w, sc1[j * 4 + 3]);
    }
  }

  // scale + key padding mask
  const v4f* kpm4 = (const v4f*)(kpm + (size_t)b * S_);
#pragma unroll
  for (int j = 0; j < 4; ++j) {
    const v4f pm = kpm4[j * 256 + t];
    sc0[j * 4 + 0] = sc0[j * 4 + 0] * SCALE_ + pm.x;
    sc0[j * 4 + 1] = sc0[j * 4 + 1] * SCALE_ + pm.y;
    sc0[j * 4 + 2] = sc0[j * 4 + 2] * SCALE_ + pm.z;
    sc0[j * 4 + 3] = sc0[j * 4 + 3] * SCALE_ + pm.w;
    sc1[j * 4 + 0] = sc1[j * 4 + 0] * SCALE_ + pm.x;
    sc1[j * 4 + 1] = sc1[j * 4 + 1] * SCALE_ + pm.y;
    sc1[j * 4 + 2] = sc1[j * 4 + 2] * SCALE_ + pm.z;
    sc1[j * 4 + 3] = sc1[j * 4 + 3] * SCALE_ + pm.w;
  }

  // ---- Softmax (max, exp, sum) across the 4096 positions ----------------
  float m0 = -3.4e38f, m1 = -3.4e38f;
#pragma unroll
  for (int j = 0; j < 16; ++j) { m0 = fmaxf(m0, sc0[j]); m1 = fmaxf(m1, sc1[j]); }
  m0 = wave_max(m0); m1 = wave_max(m1);
  if (lane == 0) { red0[wid] = m0; red1[wid] = m1; }
  __syncthreads();
  m0 = red0[0]; m1 = red1[0];
#pragma unroll
  for (int w = 1; w < 8; ++w) { m0 = fmaxf(m0, red0[w]); m1 = fmaxf(m1, red1[w]); }
  __syncthreads();

  float s0 = 0.f, s1 = 0.f;
#pragma unroll
  for (int j = 0; j < 16; ++j) {
    sc0[j] = __expf(sc0[j] - m0); s0 += sc0[j];
    sc1[j] = __expf(sc1[j] - m1); s1 += sc1[j];
  }
  s0 = wave_sum(s0); s1 = wave_sum(s1);
  if (lane == 0) { red0[wid] = s0; red1[wid] = s1; }
  __syncthreads();
  s0 = 0.f; s1 = 0.f;
#pragma unroll
  for (int w = 0; w < 8; ++w) { s0 += red0[w]; s1 += red1[w]; }
  __syncthreads();

  // ---- Phase 2: V-cache blend + store (NT, b128) + attn*V ---------------
  for (int d = 0; d < HD_; ++d) {
    const size_t rowoff = ((size_t)b * DKV_ + kvh * HD_ + d) * S_;
    const v4f* src4 = (const v4f*)(vc_in + rowoff);
    v4f*       dst4 = (v4f*)(vc_out + rowoff);
    if (d + 1 < HD_) __builtin_prefetch(src4 + 1024 + t, 0, 1);
    const float vd = vvs[d];
    float a0 = 0.f, a1 = 0.f;
#pragma unroll
    for (int j = 0; j < 4; ++j) {
      const v4f o = __builtin_nontemporal_load(src4 + j * 256 + t);
      const v4f m = umr[j];
      v4f nv;
      nv.x = fmaf(vd - o.x, m.x, o.x);
      nv.y = fmaf(vd - o.y, m.y, o.y);
      nv.z = fmaf(vd - o.z, m.z, o.z);
      nv.w = fmaf(vd - o.w, m.w, o.w);
      __builtin_nontemporal_store(nv, dst4 + j * 256 + t);
      a0 = fmaf(sc0[j * 4 + 0], nv.x, a0);
      a0 = fmaf(sc0[j * 4 + 1], nv.y, a0);
      a0 = fmaf(sc0[j * 4 + 2], nv.z, a0);
      a0 = fmaf(sc0[j * 4 + 3], nv.w, a0);
      a1 = fmaf(sc1[j * 4 + 0], nv.x, a1);
      a1 = fmaf(sc1[j * 4 + 1], nv.y, a1);
      a1 = fmaf(sc1[j * 4 + 2], nv.z, a1);
      a1 = fmaf(sc1[j * 4 + 3], nv.w, a1);
    }
    a0 = wave_sum(a0); a1 = wave_sum(a1);
    if (lane == 0) { atomicAdd(&o0s[d], a0); atomicAdd(&o1s[d], a1); }
  }
  __syncthreads();
  if (t < HD_) {
    oh[(size_t)b * 2048 + h0 * HD_ + t]       = o0s[t] / s0;
    oh[(size_t)b * 2048 + (h0 + 1) * HD_ + t] = o1s[t] / s1;
  }
}

// ---------------------------------------------------------------------------
// Kernel 4: output projection. out[b][n] = sum_k oh[b][k] * Wo[n][k]
// 64 tiles of 16x16, K = 2048, WMMA f32.
// ---------------------------------------------------------------------------
__global__ __launch_bounds__(32) void out_proj_kernel(
    const float* __restrict__ oh, const float* __restrict__ Wo,
    float* __restrict__ out) {
  const int lane = threadIdx.x & 31;
  const int row  = lane & 15;
  const int hi   = lane >> 4;
  const int n0   = blockIdx.x * 16;

  const float* wrow = Wo + (size_t)(n0 + row) * 2048;
  const float* xrow = oh + (size_t)row * 2048;

  v8f c = gemm_tile_f32(xrow, wrow, 2048);

  for (int r = 0; r < 8; ++r) {
    const int m = r + 8 * hi;
    out[(size_t)m * HID_ + n0 + row] = c[r];
  }
}

// ---------------------------------------------------------------------------
extern "C" void kernel_launch(void* const* d_in, const int* in_sizes, int n_in,
                              void* d_out, int out_size, void* d_ws, size_t ws_size,
                              hipStream_t stream) {
  (void)in_sizes; (void)n_in; (void)out_size; (void)ws_size;

  const float* x    = (const float*)d_in[0];   // (B, HID)
  const float* rc   = (const float*)d_in[1];   // (B, HD)
  const float* rs   = (const float*)d_in[2];   // (B, HD)
  const float* kpm  = (const float*)d_in[3];   // (B, S)
  const float* um   = (const float*)d_in[4];   // (B, S)
  const float* kcin = (const float*)d_in[5];   // (B, DKV, 1, S)
  const float* vcin = (const float*)d_in[6];   // (B, DKV, 1, S)
  const float* Wq   = (const float*)d_in[7];   // (2048, 1024)
  const float* Wk   = (const float*)d_in[8];   // (1024, 1024)
  const float* Wv   = (const float*)d_in[9];   // (1024, 1024)
  const float* Wo   = (const float*)d_in[10];  // (1024, 2048)
  const float* qnw  = (const float*)d_in[11];  // (HD,)
  const float* knw  = (const float*)d_in[12];  // (HD,)

  float* out    = (float*)d_out;                       // (B, HID)
  float* kc_out = out + (size_t)B_ * HID_;             // (B, DKV, 1, S)
  float* vc_out = kc_out + (size_t)B_ * DKV_ * S_;     // (B, DKV, 1, S)

  float* ws  = (float*)d_ws;
  float* qkv = ws;                         // B*4096
  float* qf  = qkv + (size_t)B_ * 4096;    // B*2048 (q after norm+rope)
  float* kf  = qf  + (size_t)B_ * 2048;    // B*1024 (k after norm+rope)
  float* oh  = kf  + (size_t)B_ * 1024;    // B*2048 (attention head outputs)

  qkv_proj_kernel<<<256, 32, 0, stream>>>(x, Wq, Wk, Wv, qkv);
  norm_rope_kernel<<<B_ * 24, 128, 0, stream>>>(qkv, qnw, knw, rc, rs, qf, kf);
  attn_fused_kernel<<<B_ * NKV_, 256, 0, stream>>>(qf, kf, qkv, um, kpm,
                                                   kcin, vcin, kc_out, vc_out, oh);
  out_proj_kernel<<<64, 32, 0, stream>>>(oh, Wo, out);
}